// PointerNetwork_84146999263673
// MI455X (gfx1250) — compile-verified
//
#include <hip/hip_runtime.h>
#include <math.h>

// Problem constants
constexpr int NB = 256;   // batch
constexpr int NS = 256;   // sequence
constexpr int NH = 128;   // hidden
constexpr int NG = 512;   // 4*H
#define NEG_INF (-3.4028234663852886e38f)

typedef __attribute__((ext_vector_type(2))) float v2f;
typedef __attribute__((ext_vector_type(8))) float v8f;

// gfx1250 has V_TANH_F32 in the TRANS pipe; use it if the builtin exists.
__device__ __forceinline__ float fast_tanh(float x) {
#if defined(__gfx1250__) && __has_builtin(__builtin_amdgcn_tanhf)
  return __builtin_amdgcn_tanhf(x);
#else
  return tanhf(x);
#endif
}

// ---------------- one-time precompute kernels ----------------

// Wfused[n][j] = sum_k Wih[n][k] * Wdec[k][j]   (512x128)
__global__ void k_wfused(const float* __restrict__ Wih, const float* __restrict__ Wdec,
                         float* __restrict__ Wfused) {
  int n = blockIdx.x, j = threadIdx.x;
  float acc = 0.f;
  for (int k = 0; k < NH; ++k) acc += Wih[n * NH + k] * Wdec[k * NH + j];
  Wfused[n * NH + j] = acc;
}

// bfused[n] = sum_k Wih[n][k]*bdec[k] + bih[n] + bhh[n]
__global__ void k_bfused(const float* __restrict__ Wih, const float* __restrict__ bdec,
                         const float* __restrict__ bih, const float* __restrict__ bhh,
                         float* __restrict__ bfused) {
  int n = blockIdx.x * blockDim.x + threadIdx.x;
  float acc = bih[n] + bhh[n];
  for (int k = 0; k < NH; ++k) acc += Wih[n * NH + k] * bdec[k];
  bfused[n] = acc;
}

// M2[h][f] = sum_k Wa[h][128+k]*Wdyn[k][f];  ucol[h] = sum_k Wa[h][128+k]*bdyn[k]
__global__ void k_m2(const float* __restrict__ Wa, const float* __restrict__ Wdyn,
                     const float* __restrict__ bdyn, float* __restrict__ M2,
                     float* __restrict__ ucol) {
  int h = threadIdx.x;
  float m0 = 0.f, m1 = 0.f, uc = 0.f;
  for (int k = 0; k < NH; ++k) {
    float w = Wa[h * 384 + 128 + k];
    m0 += w * Wdyn[k * 2 + 0];
    m1 += w * Wdyn[k * 2 + 1];
    uc += w * bdyn[k];
  }
  M2[2 * h] = m0; M2[2 * h + 1] = m1; ucol[h] = uc;
}

// h=c=0, dec_in = static_emb[:,:,0], mask=1
__global__ void k_init(const float* __restrict__ SE, float* __restrict__ hbuf,
                       float* __restrict__ cbuf, float* __restrict__ dec_in,
                       float* __restrict__ maskbuf) {
  int b = blockIdx.x, tid = threadIdx.x;
  if (tid < NH) {
    hbuf[b * NH + tid] = 0.f;
    cbuf[b * NH + tid] = 0.f;
    dec_in[b * NH + tid] = SE[(size_t)b * NH * NS + (size_t)tid * NS + 0];
  }
  maskbuf[b * NS + tid] = 1.f;
}

// Pstat[b] = Wa_s(128x128) @ SE[b](128x256)  via V_WMMA_F32_16X16X4_F32
// Pcomb[b,h,s] = Pstat + M2[h,0]*dyn[b,0,s] + M2[h,1]*dyn[b,1,s] + ucol[h]
__global__ void k_precompute(const float* __restrict__ SE, const float* __restrict__ dyn,
                             const float* __restrict__ Wa, const float* __restrict__ M2,
                             const float* __restrict__ ucol, float* __restrict__ Pstat,
                             float* __restrict__ Pcomb) {
  int b = blockIdx.x;
  int wid = threadIdx.x >> 5;
  int lane = threadIdx.x & 31;
  int tile = blockIdx.y * 4 + wid;   // 0..127
  int mt = tile >> 4;                // 0..7  (h tile)
  int nt = tile & 15;                // 0..15 (s tile)
  int lm = lane & 15;
  int kh = lane >> 4;                // K-pair select
  const float* Bm = SE + (size_t)b * NH * NS;
  int row_a = (mt * 16 + lm) * 384;  // Wa row (static part cols 0..127)
  int col = nt * 16 + lm;
  v8f c = {};
  for (int k0 = 0; k0 < NH; k0 += 4) {
    v2f a, bb;
    a.x = Wa[row_a + k0 + 2 * kh];
    a.y = Wa[row_a + k0 + 2 * kh + 1];
    bb.x = Bm[(size_t)(k0 + 2 * kh) * NS + col];
    bb.y = Bm[(size_t)(k0 + 2 * kh + 1) * NS + col];
    c = __builtin_amdgcn_wmma_f32_16x16x4_f32(false, a, false, bb, (short)0, c, false, false);
  }
  float d0 = dyn[(size_t)b * 2 * NS + col];
  float d1 = dyn[(size_t)b * 2 * NS + NS + col];
  for (int r = 0; r < 8; ++r) {
    int row = mt * 16 + 8 * kh + r;
    float ps = c[r];
    float pc = ps + M2[2 * row] * d0 + M2[2 * row + 1] * d1 + ucol[row];
    size_t o = ((size_t)b * NH + row) * NS + col;
    Pstat[o] = ps;
    Pcomb[o] = pc;
  }
}

// ---------------- per-step kernels ----------------

// xg[b, 0:128]  = dec_in @ Wdec^T + bdec
// xg[b, 128:640]= dec_in @ Wfused^T + h @ Whh^T + bfused
__global__ void k_gemm_step(const float* __restrict__ dec_in, const float* __restrict__ hbuf,
                            const float* __restrict__ Wdec, const float* __restrict__ Wfused,
                            const float* __restrict__ Whh, const float* __restrict__ bdec,
                            const float* __restrict__ bfused, float* __restrict__ xg) {
  int wid = threadIdx.x >> 5, lane = threadIdx.x & 31;
  int tile = blockIdx.x * 4 + wid;   // 0..639
  int mt = tile / 40, nt = tile % 40;
  int lm = lane & 15, kh = lane >> 4;
  int n0 = nt * 16;
  int col = n0 + lm;
  int arow = (mt * 16 + lm) * NH;
  const float* W = (n0 < NH) ? (Wdec + (size_t)col * NH) : (Wfused + (size_t)(col - NH) * NH);
  v8f c = {};
  for (int k0 = 0; k0 < NH; k0 += 4) {
    v2f a, bb;
    a.x = dec_in[arow + k0 + 2 * kh];
    a.y = dec_in[arow + k0 + 2 * kh + 1];
    bb.x = W[k0 + 2 * kh];
    bb.y = W[k0 + 2 * kh + 1];
    c = __builtin_amdgcn_wmma_f32_16x16x4_f32(false, a, false, bb, (short)0, c, false, false);
  }
  if (n0 >= NH) {  // uniform per wave
    const float* W2 = Whh + (size_t)(col - NH) * NH;
    for (int k0 = 0; k0 < NH; k0 += 4) {
      v2f a, bb;
      a.x = hbuf[arow + k0 + 2 * kh];
      a.y = hbuf[arow + k0 + 2 * kh + 1];
      bb.x = W2[k0 + 2 * kh];
      bb.y = W2[k0 + 2 * kh + 1];
      c = __builtin_amdgcn_wmma_f32_16x16x4_f32(false, a, false, bb, (short)0, c, false, false);
    }
  }
  float bias = (n0 < NH) ? bdec[col] : bfused[col - NH];
  for (int r = 0; r < 8; ++r) {
    int row = mt * 16 + 8 * kh + r;
    xg[row * 640 + col] = c[r] + bias;
  }
}

// LSTM cell elementwise + q[b,:] = Wa_h @ h_new
__global__ void k_lstm(const float* __restrict__ xg, const float* __restrict__ Wa,
                       float* __restrict__ hbuf, float* __restrict__ cbuf,
                       float* __restrict__ dec_in, float* __restrict__ q) {
  int b = blockIdx.x, j = threadIdx.x;  // 128 threads
  __shared__ float hsh[NH];
  const float* row = xg + b * 640;
  float x  = row[j];
  float ig = row[128 + j], fg = row[256 + j], gg = row[384 + j], og = row[512 + j];
  float cprev = cbuf[b * NH + j];
  float si = 1.f / (1.f + expf(-ig));
  float sf = 1.f / (1.f + expf(-fg));
  float so = 1.f / (1.f + expf(-og));
  float cn = sf * cprev + si * fast_tanh(gg);
  float hn = so * fast_tanh(cn);
  cbuf[b * NH + j] = cn;
  hbuf[b * NH + j] = hn;
  dec_in[b * NH + j] = x;     // next step's decoder input
  hsh[j] = hn;
  __syncthreads();
  float acc = 0.f;
  for (int k = 0; k < NH; ++k) acc += Wa[j * 384 + 256 + k] * hsh[k];
  q[b * NH + j] = acc;
}

__device__ __forceinline__ void threefry2x32(unsigned k0, unsigned k1, unsigned x0,
                                             unsigned x1, unsigned& o0, unsigned& o1) {
  unsigned ks[3] = {k0, k1, k0 ^ k1 ^ 0x1BD11BDAu};
  const int R0[4] = {13, 15, 26, 6}, R1[4] = {17, 29, 16, 24};
  x0 += ks[0]; x1 += ks[1];
#pragma unroll
  for (int j = 0; j < 5; ++j) {
    const int* R = (j & 1) ? R1 : R0;
#pragma unroll
    for (int r = 0; r < 4; ++r) {
      x0 += x1;
      x1 = (x1 << R[r]) | (x1 >> (32 - R[r]));
      x1 ^= x0;
    }
    x0 += ks[(j + 1) % 3];
    x1 += ks[(j + 2) % 3] + (unsigned)(j + 1);
  }
  o0 = x0; o1 = x1;
}

// jax iota-split counter bits -> uniform [tiny,1) -> gumbel
__device__ __forceinline__ float gumbel_for(unsigned fk0, unsigned fk1, unsigned idx) {
  const unsigned half = 32768u;
  unsigned x0 = (idx < half) ? idx : idx - half;
  unsigned x1 = (idx < half) ? idx + half : idx;
  unsigned o0, o1;
  threefry2x32(fk0, fk1, x0, x1, o0, o1);
  unsigned bits = (idx < half) ? o0 : o1;
  float u = __uint_as_float(0x3f800000u | (bits >> 9)) - 1.0f;  // [0,1)
  const float tiny = 1.1754944e-38f;
  u = u * (1.0f - tiny) + tiny;
  return -logf(-logf(u));
}

// Fused: scores (v^T tanh(P + q)) + masking + log-softmax + Gumbel argmax sample
//        + outputs + mask update + Pcomb column refresh.  One block per batch row.
__global__ void k_score_sample(float* __restrict__ Pcomb, const float* __restrict__ Pstat,
                               const float* __restrict__ q, const float* __restrict__ v_att,
                               const float* __restrict__ ucol, float* __restrict__ maskbuf,
                               float* __restrict__ out_tours, float* __restrict__ out_logps,
                               int t) {
  int b = blockIdx.x, tid = threadIdx.x;  // 128 threads, 2 columns each
  int s0 = tid * 2;
  const float* Pb = Pcomb + (size_t)b * NH * NS;

  __shared__ float vsh[NH];
  __shared__ float qsh[NH];
  __shared__ float slog[NS];
  __shared__ float sred[NH];
  __shared__ float sval[NH];
  __shared__ int   sidx[NH];
  __shared__ int   sptr;

  vsh[tid] = v_att[tid];
  qsh[tid] = q[b * NH + tid];
  __syncthreads();

  float acc0 = 0.f, acc1 = 0.f;
#pragma unroll 4
  for (int h = 0; h < NH; ++h) {
    float2 p = *(const float2*)(Pb + (size_t)h * NS + s0);
    float qh = qsh[h], vh = vsh[h];
    acc0 += vh * fast_tanh(p.x + qh);
    acc1 += vh * fast_tanh(p.y + qh);
  }
  bool ok0 = (t == 0) ? (s0 == 0) : (maskbuf[b * NS + s0] > 0.f);
  bool ok1 = (t == 0) ? false : (maskbuf[b * NS + s0 + 1] > 0.f);
  float lg0 = ok0 ? acc0 : NEG_INF;
  float lg1 = ok1 ? acc1 : NEG_INF;
  slog[s0] = lg0; slog[s0 + 1] = lg1;

  // max
  sred[tid] = fmaxf(lg0, lg1); __syncthreads();
  for (int off = 64; off > 0; off >>= 1) {
    if (tid < off) sred[tid] = fmaxf(sred[tid], sred[tid + off]);
    __syncthreads();
  }
  float m = sred[0]; __syncthreads();
  // sum exp
  sred[tid] = expf(lg0 - m) + expf(lg1 - m); __syncthreads();
  for (int off = 64; off > 0; off >>= 1) {
    if (tid < off) sred[tid] += sred[tid + off];
    __syncthreads();
  }
  float lse = m + logf(sred[0]); __syncthreads();

  // Gumbel perturbation (jax.random.fold_in(key(42), t) + categorical)
  unsigned fk0, fk1;
  threefry2x32(0u, 42u, 0u, (unsigned)t, fk0, fk1);
  float z0 = lg0 + gumbel_for(fk0, fk1, (unsigned)(b * NS + s0));
  float z1 = lg1 + gumbel_for(fk0, fk1, (unsigned)(b * NS + s0 + 1));
  float zb; int ib;
  if (z0 >= z1) { zb = z0; ib = s0; } else { zb = z1; ib = s0 + 1; }
  sval[tid] = zb; sidx[tid] = ib; __syncthreads();
  for (int off = 64; off > 0; off >>= 1) {
    if (tid < off) {
      float v2 = sval[tid + off]; int i2 = sidx[tid + off];
      if (v2 > sval[tid] || (v2 == sval[tid] && i2 < sidx[tid])) {
        sval[tid] = v2; sidx[tid] = i2;
      }
    }
    __syncthreads();
  }
  if (tid == 0) {
    sptr = sidx[0];
    out_tours[b * NS + t] = (float)sptr;
    out_logps[b * NS + t] = slog[sptr] - lse;
    maskbuf[b * NS + sptr] = 0.f;
  }
  __syncthreads();
  int ptr = sptr;
  // selected position's dyn column is now zero -> Pcomb col = Pstat col + ucol
  size_t o = ((size_t)b * NH + tid) * NS + ptr;
  Pcomb[o] = Pstat[o] + ucol[tid];
}

// ---------------- launcher ----------------

extern "C" void kernel_launch(void* const* d_in, const int* in_sizes, int n_in,
                              void* d_out, int out_size, void* d_ws, size_t ws_size,
                              hipStream_t stream) {
  (void)in_sizes; (void)n_in; (void)out_size; (void)ws_size;
  const float* SE    = (const float*)d_in[1];   // static_embeddings (B,H,S)
  const float* dyn   = (const float*)d_in[2];   // dynamic (B,F,S)
  const float* Wdyn  = (const float*)d_in[3];
  const float* bdyn  = (const float*)d_in[4];
  const float* Wdec  = (const float*)d_in[5];
  const float* bdec  = (const float*)d_in[6];
  const float* Wih   = (const float*)d_in[7];
  const float* Whh   = (const float*)d_in[8];
  const float* bih   = (const float*)d_in[9];
  const float* bhh   = (const float*)d_in[10];
  const float* Wa    = (const float*)d_in[11];
  const float* v_att = (const float*)d_in[12];
  float* out = (float*)d_out;                   // [tours (B,S) | logps (B,S)]
  float* ws = (float*)d_ws;

  constexpr size_t PHS = (size_t)NB * NH * NS;  // 8388608
  float* Pcomb  = ws;
  float* Pstat  = ws + PHS;
  float* xg     = ws + 2 * PHS;                 // B*640
  float* hbuf   = xg + (size_t)NB * 640;
  float* cbuf   = hbuf + (size_t)NB * NH;
  float* decin  = cbuf + (size_t)NB * NH;
  float* q      = decin + (size_t)NB * NH;
  float* maskb  = q + (size_t)NB * NH;
  float* Wfused = maskb + (size_t)NB * NS;
  float* bfused = Wfused + (size_t)NG * NH;
  float* M2     = bfused + NG;
  float* ucol   = M2 + 2 * NH;

  k_wfused<<<NG, NH, 0, stream>>>(Wih, Wdec, Wfused);
  k_bfused<<<4, 128, 0, stream>>>(Wih, bdec, bih, bhh, bfused);
  k_m2<<<1, NH, 0, stream>>>(Wa, Wdyn, bdyn, M2, ucol);
  k_init<<<NB, NS, 0, stream>>>(SE, hbuf, cbuf, decin, maskb);
  dim3 gA(NB, 32);
  k_precompute<<<gA, 128, 0, stream>>>(SE, dyn, Wa, M2, ucol, Pstat, Pcomb);

  for (int t = 0; t < NS; ++t) {
    k_gemm_step<<<160, 128, 0, stream>>>(decin, hbuf, Wdec, Wfused, Whh, bdec, bfused, xg);
    k_lstm<<<NB, NH, 0, stream>>>(xg, Wa, hbuf, cbuf, decin, q);
    k_score_sample<<<NB, NH, 0, stream>>>(Pcomb, Pstat, q, v_att, ucol, maskb,
                                          out, out + (size_t)NB * NS, t);
  }
}